// FlashHSA_Inference_15547781612182
// MI455X (gfx1250) — compile-verified
//
#include <hip/hip_runtime.h>

#define B_ 16
#define T_ 4096
#define HKV_ 8
#define HD_ 128
#define DM_ 4096
#define HQ_ 32
#define CHUNK_ 64
#define TOPK_ 16
#define WINDOW_ 512
#define C_ 64
#define G_ 4
#define KSPLIT 8
#define KLEN (DM_ / KSPLIT) /* 512 */

typedef float v2f __attribute__((ext_vector_type(2)));
typedef float v4f __attribute__((ext_vector_type(4)));
typedef float v8f __attribute__((ext_vector_type(8)));

// ---------------------------------------------------------------------------
// Kernel 1/5: Y_part[ks] = X[16,4096] @ W[4096,4096] (K-split by 8)
// One wave = one 16x16 output tile, V_WMMA_F32_16X16X4_F32 over K.
// A layout (ISA 7.12.2, 32-bit A 16x4): lanes0-15 -> K0,K1 ; lanes16-31 -> K2,K3
// B layout (4x16, row striped across lanes): vgpr0 = rows K0/K2, vgpr1 = K1/K3
// D layout (32-bit C/D 16x16): vgpr r -> M=r (lanes0-15), M=r+8 (lanes16-31)
// ---------------------------------------------------------------------------
__global__ __launch_bounds__(128) void gemm16_wmma(
    const float* __restrict__ X, const float* __restrict__ W,
    float* __restrict__ part)
{
  const int lane = threadIdx.x & 31;
  const int wv   = threadIdx.x >> 5;
  const int nl   = lane & 15;
  const int hi   = lane >> 4;
  const int m    = nl;
  const int n0   = blockIdx.x * 64 + wv * 16;
  const int ks   = blockIdx.y;
  const int k0   = ks * KLEN;

  const float* xp = X + (size_t)m * DM_ + k0 + 2 * hi;
  const float* wp = W + (size_t)(k0 + 2 * hi) * DM_ + n0 + nl;

  v8f acc = {};
#pragma unroll 4
  for (int kk = 0; kk < KLEN; kk += 4) {
    v2f a = *(const v2f*)xp;          // A[m][k+2hi], A[m][k+2hi+1]
    v2f b;
    b.x = wp[0];                      // W[k+2hi][n0+nl]
    b.y = wp[DM_];                    // W[k+2hi+1][n0+nl]
    acc = __builtin_amdgcn_wmma_f32_16x16x4_f32(
        false, a, false, b, (short)0, acc, false, false);
    xp += 4;
    wp += (size_t)4 * DM_;
  }

  float* op = part + (size_t)ks * 16 * DM_ + n0 + nl;
#pragma unroll
  for (int r = 0; r < 8; ++r) {
    int mo = r + 8 * hi;
    op[(size_t)mo * DM_] = acc[r];
  }
}

// ---------------------------------------------------------------------------
// Kernel 2: reduce K-split partials + bias, per-head RMSNorm -> qn [16,4096]
// One wave per (b,h) row of 128 elements.
// ---------------------------------------------------------------------------
__global__ __launch_bounds__(256) void rms_q(
    const float* __restrict__ part, const float* __restrict__ bq,
    const float* __restrict__ qn_w, float* __restrict__ qn)
{
  const int lane = threadIdx.x & 31;
  const int row  = blockIdx.x * 8 + (threadIdx.x >> 5); // 0..511
  const int b = row >> 5;
  const int h = row & 31;
  const int col = h * HD_ + lane * 4;

  float v[4];
#pragma unroll
  for (int j = 0; j < 4; ++j) {
    float s = bq[col + j];
#pragma unroll
    for (int ks = 0; ks < KSPLIT; ++ks)
      s += part[((size_t)ks * 16 + b) * DM_ + col + j];
    v[j] = s;
  }
  float ss = v[0]*v[0] + v[1]*v[1] + v[2]*v[2] + v[3]*v[3];
#pragma unroll
  for (int msk = 1; msk < 32; msk <<= 1) ss += __shfl_xor(ss, msk, 32);
  float rn = rsqrtf(ss / (float)HD_ + 1e-6f);
#pragma unroll
  for (int j = 0; j < 4; ++j)
    qn[(size_t)b * DM_ + col + j] = qn_w[lane * 4 + j] * v[j] * rn;
}

// ---------------------------------------------------------------------------
// Kernel 3: landmark scores + validity + top-16 + softmax weights
// One block per (b, kvh); 4 waves = 4 GQA query heads.
// ---------------------------------------------------------------------------
__global__ __launch_bounds__(128) void lmk_topk(
    const float* __restrict__ qn, const float* __restrict__ kc,
    const float* __restrict__ lmkn_w, const float* __restrict__ lse,
    const int* __restrict__ seq_lens,
    int* __restrict__ top_i, float* __restrict__ cw)
{
  __shared__ __align__(16) float qlm[G_][HD_];
  __shared__ float s_l[G_][C_];
  __shared__ float topv[G_][TOPK_];
  __shared__ int   topix[G_][TOPK_];

  const int b   = blockIdx.x >> 3;
  const int kvh = blockIdx.x & 7;
  const int tid  = threadIdx.x;
  const int lane = tid & 31;
  const int wv   = tid >> 5;
  const float scale = 0.08838834764831845f; // 1/sqrt(128)

  // stage q*lmkn_w for the 4 heads of this kv group
  {
    int i = tid * 4; // 0..508, covers 512 = G_*HD_
#pragma unroll
    for (int j = 0; j < 4; ++j) {
      int g = (i + j) >> 7, d = (i + j) & 127;
      qlm[g][d] = qn[(size_t)b * DM_ + (kvh * G_ + g) * HD_ + d] * lmkn_w[d];
    }
  }
  __syncthreads();

  // landmark RMS + dots: wave handles 16 chunks, 2 lanes per chunk
  const int c    = wv * 16 + (lane >> 1);
  const int half = lane & 1;
  const float* kr =
      kc + (((size_t)b * T_ + (size_t)c * CHUNK_) * HKV_ + kvh) * HD_ + half * 64;
  v4f kv[16];
  float ssq = 0.f;
#pragma unroll
  for (int j = 0; j < 16; ++j) {
    kv[j] = *(const v4f*)(kr + j * 4);
    ssq += kv[j].x*kv[j].x + kv[j].y*kv[j].y + kv[j].z*kv[j].z + kv[j].w*kv[j].w;
  }
  ssq += __shfl_xor(ssq, 1, 32);
  const float rn = rsqrtf(ssq / (float)HD_ + 1e-6f);
#pragma unroll
  for (int g = 0; g < G_; ++g) {
    float dot = 0.f;
#pragma unroll
    for (int j = 0; j < 16; ++j) {
      const float* qp = &qlm[g][half * 64 + j * 4];
      dot += qp[0]*kv[j].x + qp[1]*kv[j].y + qp[2]*kv[j].z + qp[3]*kv[j].w;
    }
    dot += __shfl_xor(dot, 1, 32);
    if (half == 0) s_l[g][c] = dot * rn * scale;
  }
  __syncthreads();

  // top-16 per head: wave wv owns head g=wv
  const int g  = wv;
  const int h  = kvh * G_ + g;
  const int sl = seq_lens[b];
  const int c0 = lane, c1 = lane + 32;
  float sv0, sv1;
  {
    int cs0 = c0 * CHUNK_, cs1 = c1 * CHUNK_;
    bool v0 = (cs0 + CHUNK_ <= sl) && (cs0 < sl - WINDOW_);
    bool v1 = (cs1 + CHUNK_ <= sl) && (cs1 < sl - WINDOW_);
    sv0 = v0 ? s_l[g][c0] : -1e9f;
    sv1 = v1 ? s_l[g][c1] : -1e9f;
  }
  for (int it = 0; it < TOPK_; ++it) {
    float mv; int mi;
    if (sv0 >= sv1) { mv = sv0; mi = c0; } else { mv = sv1; mi = c1; }
#pragma unroll
    for (int msk = 1; msk < 32; msk <<= 1) {
      float ov = __shfl_xor(mv, msk, 32);
      int   oi = __shfl_xor(mi, msk, 32);
      if (ov > mv || (ov == mv && oi < mi)) { mv = ov; mi = oi; }
    }
    if (lane == 0) { topv[g][it] = mv; topix[g][it] = mi; }
    if (mi == c0) sv0 = -1e30f;
    if (mi == c1) sv1 = -1e30f;
  }
  __syncthreads();

  // softmax over [top16 scores, lse_swa]
  float val = -1e30f;
  if (lane < TOPK_)        val = topv[g][lane];
  else if (lane == TOPK_)  val = lse[b * HQ_ + h];
  float mx = val;
#pragma unroll
  for (int msk = 1; msk < 32; msk <<= 1) mx = fmaxf(mx, __shfl_xor(mx, msk, 32));
  float e = (lane <= TOPK_) ? __expf(val - mx) : 0.f;
  float sum = e;
#pragma unroll
  for (int msk = 1; msk < 32; msk <<= 1) sum += __shfl_xor(sum, msk, 32);
  float wgt = e / fmaxf(sum, 1e-20f);
  if (lane <= TOPK_) cw[((size_t)b * HQ_ + h) * (TOPK_ + 1) + lane] = wgt;
  if (lane <  TOPK_) top_i[((size_t)b * HQ_ + h) * TOPK_ + lane] = topix[g][lane];
}

// ---------------------------------------------------------------------------
// Kernel 4: gathered per-chunk attention + weighted combine + SWA branch
// One block per (b,h); 8 waves x 2 chunk-slots; prefetch of slot-1 KV tiles
// overlaps slot-0 compute (global_prefetch_b8 path on gfx1250).
// ---------------------------------------------------------------------------
__global__ __launch_bounds__(256) void hsa_attn(
    const float* __restrict__ qn, const float* __restrict__ kc,
    const float* __restrict__ vc, const int* __restrict__ top_i,
    const float* __restrict__ cw, const int* __restrict__ seq_lens,
    float* __restrict__ o)
{
  __shared__ __align__(16) float q_l[HD_];
  __shared__ float p_l[8][CHUNK_];
  __shared__ float acc_l[8][HD_];

  const int bh = blockIdx.x;
  const int b = bh >> 5;
  const int h = bh & 31;
  const int kvh = h >> 2;
  const int tid  = threadIdx.x;
  const int lane = tid & 31;
  const int wv   = tid >> 5;
  const float scale = 0.08838834764831845f;

  if (tid < HD_) q_l[tid] = qn[(size_t)b * DM_ + h * HD_ + tid];
  for (int i = tid; i < 8 * HD_; i += 256) ((float*)acc_l)[i] = 0.f;
  __syncthreads();

  const int sl = seq_lens[b];
  const size_t row_stride = (size_t)HKV_ * HD_;
  const int t0 = lane * 2;

  // prefetch slot-1 chunk while slot-0 streams (indices already known)
  {
    const int ci1 = top_i[((size_t)b * HQ_ + h) * TOPK_ + 8 + wv];
    const float* kpre = kc + (((size_t)b * T_ + ci1 * CHUNK_ + t0) * HKV_ + kvh) * HD_;
    const float* vpre = vc + (((size_t)b * T_ + ci1 * CHUNK_ + t0) * HKV_ + kvh) * HD_;
    __builtin_prefetch(kpre, 0, 3);
    __builtin_prefetch(kpre + 16, 0, 3);
    __builtin_prefetch(vpre, 0, 3);
    __builtin_prefetch(vpre + 16, 0, 3);
  }

  for (int slot = 0; slot < 2; ++slot) {
    const int kidx = slot * 8 + wv;
    const int ci   = top_i[((size_t)b * HQ_ + h) * TOPK_ + kidx];
    const float wgt = cw[((size_t)b * HQ_ + h) * (TOPK_ + 1) + kidx];
    const int p0 = ci * CHUNK_;

    // scores for 2 tokens per lane
    const float* k0p = kc + (((size_t)b * T_ + p0 + t0) * HKV_ + kvh) * HD_;
    const float* k1p = k0p + row_stride;
    float s0 = 0.f, s1 = 0.f;
#pragma unroll 8
    for (int j = 0; j < HD_; j += 4) {
      v4f q4 = *(const v4f*)&q_l[j];
      v4f a4 = *(const v4f*)(k0p + j);
      v4f b4 = *(const v4f*)(k1p + j);
      s0 += q4.x*a4.x + q4.y*a4.y + q4.z*a4.z + q4.w*a4.w;
      s1 += q4.x*b4.x + q4.y*b4.y + q4.z*b4.z + q4.w*b4.w;
    }
    s0 = (p0 + t0     < sl) ? s0 * scale : -1e9f;
    s1 = (p0 + t0 + 1 < sl) ? s1 * scale : -1e9f;

    // chunk softmax over 64 tokens
    float mx = fmaxf(s0, s1);
#pragma unroll
    for (int msk = 1; msk < 32; msk <<= 1) mx = fmaxf(mx, __shfl_xor(mx, msk, 32));
    float e0 = __expf(s0 - mx), e1 = __expf(s1 - mx);
    float sum = e0 + e1;
#pragma unroll
    for (int msk = 1; msk < 32; msk <<= 1) sum += __shfl_xor(sum, msk, 32);
    float inv = 1.f / fmaxf(sum, 1e-20f);
    p_l[wv][t0]     = e0 * inv;
    p_l[wv][t0 + 1] = e1 * inv;
    __syncthreads();

    // P @ V : lane owns dims [lane*4, lane*4+4) -- fully coalesced row loads
    const int dd = lane * 4;
    v4f vacc = {};
    const float* vp = vc + (((size_t)b * T_ + p0) * HKV_ + kvh) * HD_ + dd;
#pragma unroll 4
    for (int t = 0; t < CHUNK_; ++t) {
      float pv = p_l[wv][t];
      v4f v4 = *(const v4f*)(vp + (size_t)t * row_stride);
      vacc.x += pv * v4.x; vacc.y += pv * v4.y;
      vacc.z += pv * v4.z; vacc.w += pv * v4.w;
    }
    acc_l[wv][dd + 0] += wgt * vacc.x;
    acc_l[wv][dd + 1] += wgt * vacc.y;
    acc_l[wv][dd + 2] += wgt * vacc.z;
    acc_l[wv][dd + 3] += wgt * vacc.w;
    __syncthreads();
  }

  // combine 8 wave partials + SWA branch (q * cw[16])
  if (tid < HD_) {
    float t = 0.f;
#pragma unroll
    for (int w2 = 0; w2 < 8; ++w2) t += acc_l[w2][tid];
    t += q_l[tid] * cw[((size_t)b * HQ_ + h) * (TOPK_ + 1) + TOPK_];
    o[(size_t)b * DM_ + h * HD_ + tid] = t;
  }
}

// ---------------------------------------------------------------------------
// Kernel 6: sum K-split partials of output GEMM + bias -> d_out
// ---------------------------------------------------------------------------
__global__ __launch_bounds__(256) void finalize_o(
    const float* __restrict__ part, const float* __restrict__ bo,
    float* __restrict__ out)
{
  int i = blockIdx.x * 256 + threadIdx.x; // 0..65535
  float s = bo[i & (DM_ - 1)];
#pragma unroll
  for (int ks = 0; ks < KSPLIT; ++ks)
    s += part[(size_t)ks * 16 * DM_ + i];
  out[i] = s;
}

// ---------------------------------------------------------------------------
extern "C" void kernel_launch(void* const* d_in, const int* in_sizes, int n_in,
                              void* d_out, int out_size, void* d_ws, size_t ws_size,
                              hipStream_t stream) {
  const float* hs   = (const float*)d_in[0];   // [16,1,4096]
  const float* kc   = (const float*)d_in[1];   // [16,4096,8,128]
  const float* vc   = (const float*)d_in[2];   // [16,4096,8,128]
  const float* lse  = (const float*)d_in[3];   // [16,32]
  const float* Wq   = (const float*)d_in[4];   // [4096,4096]
  const float* bq   = (const float*)d_in[5];   // [4096]
  const float* Wo   = (const float*)d_in[6];   // [4096,4096]
  const float* bo   = (const float*)d_in[7];   // [4096]
  const float* qnw  = (const float*)d_in[8];   // [128]
  const float* lmkn = (const float*)d_in[9];   // [128]
  const int*   sl   = (const int*)d_in[10];    // [16]
  float* out = (float*)d_out;                  // [16,1,4096] f32

  float* ws    = (float*)d_ws;
  float* part1 = ws;                                  // 8*16*4096
  float* qn    = part1 + (size_t)KSPLIT * 16 * DM_;   // 16*4096
  float* cwv   = qn + (size_t)16 * DM_;               // 16*32*17
  int*   topi  = (int*)(cwv + (size_t)B_ * HQ_ * (TOPK_ + 1)); // 16*32*16
  float* o_ws  = (float*)(topi + (size_t)B_ * HQ_ * TOPK_);    // 16*4096
  float* part2 = o_ws + (size_t)16 * DM_;             // 8*16*4096

  gemm16_wmma<<<dim3(64, KSPLIT), 128, 0, stream>>>(hs, Wq, part1);
  rms_q     <<<64, 256, 0, stream>>>(part1, bq, qnw, qn);
  lmk_topk  <<<B_ * HKV_, 128, 0, stream>>>(qn, kc, lmkn, lse, sl, topi, cwv);
  hsa_attn  <<<B_ * HQ_, 256, 0, stream>>>(qn, kc, vc, topi, cwv, sl, o_ws);
  gemm16_wmma<<<dim3(64, KSPLIT), 128, 0, stream>>>(o_ws, Wo, part2);
  finalize_o<<<256, 256, 0, stream>>>(part2, bo, out);
}